// newQConv1D_34565896798388
// MI455X (gfx1250) — compile-verified
//
#include <hip/hip_runtime.h>

// out[b, ch, pos] = cos(theta[ch,0]) * EZ(b,pos) - sin(theta[ch,0]) * EX(b,pos)
//   EZ = prod_{q=1..9} cos(col_q)  (col = [x0[pos..pos+4], x1[pos..pos+4]])
//   EX = sin(x0[pos]) * sin(x0[pos+1])
// Derived by algebraically folding the product-state encoding, the GF(2)-linear
// CNOT-ring permutation, and the RY layer (only theta[ch,0] survives under Z_0).
// The channel combine is a (positions x 2) @ (2 x 8) GEMM -> V_WMMA_F32_16X16X4_F32.

typedef __attribute__((ext_vector_type(2))) float v2f;
typedef __attribute__((ext_vector_type(8))) float v8f;

#define OUT_CH  8
#define OUT_LEN 1020
#define NPOS    (16 * OUT_LEN)   // 16320 patch positions
#define NTILES  (NPOS / 16)      // 1020 WMMA M-tiles (exact)

__global__ __launch_bounds__(128) void qconv1d_wmma_kernel(
    const float* __restrict__ x,       // (16, 2, 1024) f32
    const float* __restrict__ thetas,  // (8, 10) f32
    float* __restrict__ out)           // (16, 8, 1020) f32
{
    const int lane = threadIdx.x & 31;
    const int wave = threadIdx.x >> 5;                 // 4 waves per block
    const int tile = blockIdx.x * 4 + wave;            // 0..1019, no guard needed

    // ---- A fragment: 16x4 f32, lanes 0-15 carry K=0 (EZ) and K=1 (EX) ----
    const int m   = lane & 15;                         // M within tile == N (channel)
    const int g   = tile * 16 + m;                     // global patch position
    const int b   = g / OUT_LEN;                       // single division per lane
    const int pos = g - b * OUT_LEN;
    const float* x0 = x + (b * 2 + 0) * 1024 + pos;
    const float* x1 = x + (b * 2 + 1) * 1024 + pos;

    float s1, c1;
    sincosf(x0[1], &s1, &c1);                          // both needed: c1 in EZ, s1 in EX
    const float c2 = cosf(x0[2]), c3 = cosf(x0[3]), c4 = cosf(x0[4]);
    const float d0 = cosf(x1[0]), d1 = cosf(x1[1]), d2 = cosf(x1[2]);
    const float d3 = cosf(x1[3]), d4 = cosf(x1[4]);
    const float ez = ((c1 * c2) * (c3 * c4)) * (((d0 * d1) * (d2 * d3)) * d4);
    const float ex = sinf(x0[0]) * s1;

    const bool lo = (lane < 16);                       // upper lanes hold K=2,3 == 0
    v2f a;
    a.x = lo ? ez : 0.0f;
    a.y = lo ? ex : 0.0f;

    // ---- B fragment: 4x16 f32, row K=0 = cos(theta), K=1 = -sin(theta) ----
    const float th = (m < OUT_CH) ? thetas[m * 10] : 0.0f;   // theta[ch, 0]
    const bool  bv = lo && (m < OUT_CH);
    v2f bm;
    bm.x = bv ?  cosf(th) : 0.0f;
    bm.y = bv ? -sinf(th) : 0.0f;

    // ---- D = A x B (+0): exact f32 WMMA; EXEC is all-ones here ----
    v8f acc = {};
    v8f dmat = __builtin_amdgcn_wmma_f32_16x16x4_f32(
        /*neg_a=*/false, a, /*neg_b=*/false, bm,
        /*c_mod=*/(short)0, acc, /*reuse_a=*/false, /*reuse_b=*/false);

    // ---- Scatter D: VGPR r, lane L -> position tile*16 + (L<16 ? r : r+8),
    //      channel L&15 (only channels 0..7 are real). The 8 positions are
    //      consecutive, so carry (b2, p2) incrementally: one division total,
    //      +1 per step, +(7*OUT_LEN+1) on batch rollover. ----
    const int ch = m;
    if (ch < OUT_CH) {
        const int gg0 = tile * 16 + (lo ? 0 : 8);
        int b2 = gg0 / OUT_LEN;
        int p2 = gg0 - b2 * OUT_LEN;
        int idx = (b2 * OUT_CH + ch) * OUT_LEN + p2;
        #pragma unroll
        for (int r = 0; r < 8; ++r) {
            out[idx] = dmat[r];
            ++p2;
            const bool wrapb = (p2 == OUT_LEN);
            idx += wrapb ? (7 * OUT_LEN + 1) : 1;      // next pos, or (b2+1, pos=0)
            p2   = wrapb ? 0 : p2;
        }
    }
}

extern "C" void kernel_launch(void* const* d_in, const int* in_sizes, int n_in,
                              void* d_out, int out_size, void* d_ws, size_t ws_size,
                              hipStream_t stream) {
    const float* x      = (const float*)d_in[0];   // (16, 2, 1024)
    const float* thetas = (const float*)d_in[1];   // (8, 10)
    // d_in[2] (entangle, 1024x1024) is the deterministic CNOT-ring permutation;
    // it has been folded into the closed form above and is not read.
    float* out = (float*)d_out;                    // (16, 8, 1020)

    // 1020 tiles of 16 positions; 4 waves/block -> 255 blocks, every wave active
    // (EXEC all-ones at the WMMA, as the ISA requires).
    qconv1d_wmma_kernel<<<dim3(NTILES / 4), dim3(128), 0, stream>>>(x, thetas, out);
}